// CTCLayer_76038101008789
// MI455X (gfx1250) — compile-verified
//
#include <hip/hip_runtime.h>
#include <hip/hip_bf16.h>

typedef __attribute__((ext_vector_type(2))) float v2f;
typedef __attribute__((ext_vector_type(8))) float v8f;

#define Bc 64
#define Tc 512
#define Vc 1296
#define Sc 32
#define Lc 65            // 2*S+1
#define NEGC (-1e30f)

__device__ __forceinline__ float lse2(float x, float y) {
    // logaddexp, safe for -1e30 sentinels (exp underflows to 0, no NaN)
    float m = fmaxf(x, y);
    float d = fabsf(x - y);
    return m + log1pf(__expf(-d));
}

// ---------------------------------------------------------------------------
// Kernel 1: per-row softmax stats. One wave handles 16 rows.
// Pass 1: row max (lane halves + shfl_xor(16) combine).
// Pass 2: sum(exp(x-max)) via V_WMMA_F32_16X16X4_F32 with all-ones B, using
//   FOUR independent accumulators to break the WMMA->WMMA RAW chain
//   (hazard table CDNA5 ISA 7.12.1); combined with elementwise v8f adds.
//   A layout (f32 16x4): lane -> M = lane&15, K pair = (lane>>4)*2.
//   D layout: VGPR g, lanes 0-15 -> row g; lanes 16-31 -> row g+8.
// ---------------------------------------------------------------------------
__global__ void ctc_rowstats_wmma(const float* __restrict__ logits,
                                  float* __restrict__ logZ, int rows) {
    const int lane   = threadIdx.x & 31;
    const int wave   = blockIdx.x * (blockDim.x >> 5) + (threadIdx.x >> 5);
    const int row0   = wave * 16;
    if (row0 >= rows) return;                 // wave-uniform; EXEC stays all-1s
    const int m      = lane & 15;
    const int koff   = (lane >> 4) << 1;      // 0 or 2
    const float* rp  = logits + (size_t)(row0 + m) * Vc + koff;

    // ---- pass 1: max over the row (each lane covers half, interleaved) ----
    float mx = -INFINITY;
    for (int k = 0; k < Vc / 4; ++k) {
        float x0 = rp[4 * k + 0];
        float x1 = rp[4 * k + 1];
        mx = fmaxf(mx, fmaxf(x0, x1));
    }
    mx = fmaxf(mx, __shfl_xor(mx, 16, 32));   // full row max in both lane halves

    // ---- pass 2: WMMA-accumulated sum of exp(x - max), 4 parallel chains ----
    v8f acc0 = {}, acc1 = {}, acc2 = {}, acc3 = {};
    const v2f ones = {1.0f, 1.0f};
    for (int k = 0; k < Vc / 16; ++k) {       // 1296/16 = 81, exact
        const float* p = rp + 16 * k;
        v2f a0, a1, a2, a3;
        a0.x = __expf(p[0]  - mx);  a0.y = __expf(p[1]  - mx);
        a1.x = __expf(p[4]  - mx);  a1.y = __expf(p[5]  - mx);
        a2.x = __expf(p[8]  - mx);  a2.y = __expf(p[9]  - mx);
        a3.x = __expf(p[12] - mx);  a3.y = __expf(p[13] - mx);
        // (neg_a, A, neg_b, B, c_mod, C, reuse_a, reuse_b)
        acc0 = __builtin_amdgcn_wmma_f32_16x16x4_f32(false, a0, false, ones,
                                                     (short)0, acc0, false, false);
        acc1 = __builtin_amdgcn_wmma_f32_16x16x4_f32(false, a1, false, ones,
                                                     (short)0, acc1, false, false);
        acc2 = __builtin_amdgcn_wmma_f32_16x16x4_f32(false, a2, false, ones,
                                                     (short)0, acc2, false, false);
        acc3 = __builtin_amdgcn_wmma_f32_16x16x4_f32(false, a3, false, ones,
                                                     (short)0, acc3, false, false);
    }
    v8f acc = (acc0 + acc1) + (acc2 + acc3);  // elementwise: D layout matches

    // ---- write logZ = max + log(sumexp). lane0 -> rows 0..7, lane16 -> 8..15
    #pragma unroll
    for (int g = 0; g < 8; ++g) {
        float rmA = __shfl(mx, g,     32);    // max of row g
        float rmB = __shfl(mx, g + 8, 32);    // max of row g+8
        if (lane == 0)  logZ[row0 + g]     = rmA + __logf(acc[g]);
        if (lane == 16) logZ[row0 + 8 + g] = rmB + __logf(acc[g]);
    }
}

// ---------------------------------------------------------------------------
// Kernel 2: gather extended-label log-probs:
//   lp_ext[b,t,s] = logits[b,t,ext[s]] - logZ[b,t],  ext[s] = odd(s)?tgt[s/2]:0
// ---------------------------------------------------------------------------
__global__ void ctc_gather(const float* __restrict__ logits,
                           const float* __restrict__ logZ,
                           const int*   __restrict__ targets,
                           float* __restrict__ lp_ext, int n) {
    int idx = blockIdx.x * blockDim.x + threadIdx.x;
    if (idx >= n) return;
    int s = idx % Lc;
    int r = idx / Lc;                  // r = b*T + t
    int b = r / Tc;
    int lab = (s & 1) ? targets[b * Sc + (s >> 1)] : 0;   // BLANK = 0
    lp_ext[idx] = logits[(size_t)r * Vc + lab] - logZ[r];
}

// ---------------------------------------------------------------------------
// Kernel 3: CTC forward (alpha) recursion. One block per batch element.
// alpha[s] lives in LDS; 65 active lanes, sequential over t with barriers.
// ---------------------------------------------------------------------------
__global__ void ctc_alpha(const float* __restrict__ lp_ext,
                          const int*   __restrict__ targets,
                          const int*   __restrict__ in_len,
                          const int*   __restrict__ tg_len,
                          float* __restrict__ nll) {
    __shared__ float alpha[Lc];
    __shared__ int   tgt[Sc];
    const int b = blockIdx.x;
    const int s = threadIdx.x;

    if (s < Sc) tgt[s] = targets[b * Sc + s];
    __syncthreads();

    bool skip = false;                 // transition s-2 -> s allowed?
    if (s < Lc && (s & 1)) {
        int j = s >> 1;
        skip = (j == 0) || (tgt[j] != tgt[j - 1]);
    }

    const float* lp = lp_ext + (size_t)b * Tc * Lc;
    if (s < Lc) alpha[s] = (s < 2) ? lp[s] : NEGC;
    __syncthreads();

    const int len = in_len[b];
    for (int t = 1; t < len; ++t) {
        if (s == 0 && t + 1 < len)
            __builtin_prefetch(lp + (size_t)(t + 1) * Lc, 0, 0); // global_prefetch_b8
        float v = NEGC;
        if (s < Lc) {
            float a1 = alpha[s];
            float a2 = (s >= 1)         ? alpha[s - 1] : NEGC;
            float a3 = (s >= 2 && skip) ? alpha[s - 2] : NEGC;
            v = lse2(lse2(a1, a2), a3) + lp[(size_t)t * Lc + s];
        }
        __syncthreads();
        if (s < Lc) alpha[s] = v;
        __syncthreads();
    }

    if (s == 0) {
        int tl = tg_len[b];
        float le = alpha[2 * tl];
        float ll = (tl >= 1) ? alpha[2 * tl - 1] : NEGC;
        float r  = -lse2(le, ll);
        if (isinf(r) || r > 1e29f) r = 0.0f;   // zero_infinity=True
        nll[b] = r / (float)tl;
    }
}

// ---------------------------------------------------------------------------
// Kernel 4: mean over batch -> scalar output
// ---------------------------------------------------------------------------
__global__ void ctc_mean(const float* __restrict__ nll, float* __restrict__ out,
                         int n) {
    __shared__ float sh[64];
    int i = threadIdx.x;
    sh[i] = (i < n) ? nll[i] : 0.0f;
    __syncthreads();
    for (int off = 32; off > 0; off >>= 1) {
        if (i < off) sh[i] += sh[i + off];
        __syncthreads();
    }
    if (i == 0) out[0] = sh[0] / (float)n;
}

// ---------------------------------------------------------------------------
extern "C" void kernel_launch(void* const* d_in, const int* in_sizes, int n_in,
                              void* d_out, int out_size, void* d_ws, size_t ws_size,
                              hipStream_t stream) {
    const float* logits  = (const float*)d_in[0];   // (B,T,V) f32
    const int*   targets = (const int*)d_in[1];     // (B,S)   i32
    const int*   in_len  = (const int*)d_in[2];     // (B,)    i32
    const int*   tg_len  = (const int*)d_in[3];     // (B,)    i32
    float*       out     = (float*)d_out;

    const int R = Bc * Tc;                          // 32768 rows
    float* wsf    = (float*)d_ws;
    float* logZ   = wsf;                            // R floats
    float* lp_ext = wsf + R;                        // R*L floats (~8.5 MB)
    float* nllb   = lp_ext + (size_t)R * Lc;        // B floats

    // K1: 16 rows per wave, 8 waves per block -> 128 rows/block
    int k1_blocks = (R / 16 + 7) / 8;
    ctc_rowstats_wmma<<<k1_blocks, 256, 0, stream>>>(logits, logZ, R);

    // K2: gather lp_ext
    int n2 = R * Lc;
    ctc_gather<<<(n2 + 255) / 256, 256, 0, stream>>>(logits, logZ, targets,
                                                     lp_ext, n2);

    // K3: alpha recursion, one block per batch element (65 active lanes)
    ctc_alpha<<<Bc, 96, 0, stream>>>(lp_ext, targets, in_len, tg_len, nllb);

    // K4: batch mean -> scalar
    ctc_mean<<<1, 64, 0, stream>>>(nllb, out, Bc);
}